// Net_41068477285094
// MI455X (gfx1250) — compile-verified
//
#include <hip/hip_runtime.h>

typedef float v2f __attribute__((ext_vector_type(2)));
typedef float v8f __attribute__((ext_vector_type(8)));

// =====================================================================
// Device-side Clebsch-Gordan computation (exact mirror of reference)
// =====================================================================
__device__ const int PATH_I[15] = {0,0,1,1,1,1,0,1,1,2,2,2,2,2,2};
__device__ const int PATH_F[15] = {0,1,0,1,1,1,2,2,2,2,2,2,0,1,1};
__device__ const int PATH_O[15] = {0,1,1,0,1,2,2,1,2,0,1,2,2,1,2};

struct cdbl { double re, im; };
__device__ __forceinline__ cdbl cmul(cdbl a, cdbl b) {
    cdbl r; r.re = a.re*b.re - a.im*b.im; r.im = a.re*b.im + a.im*b.re; return r;
}
__device__ double factd(int n) { double r = 1.0; for (int i = 2; i <= n; ++i) r *= (double)i; return r; }

__device__ double su2_elem(int j1, int j2, int j3, int m1, int m2) {
    int m3 = m1 + m2;
    if (m3 < -j3 || m3 > j3) return 0.0;
    double pref = sqrt((double)(2*j3+1) * factd(j3+j1-j2)*factd(j3-j1+j2)*factd(j1+j2-j3) / factd(j1+j2+j3+1));
    pref *= sqrt(factd(j3+m3)*factd(j3-m3)*factd(j1-m1)*factd(j1+m1)*factd(j2-m2)*factd(j2+m2));
    double s = 0.0;
    for (int k = 0; k <= j1+j2-j3; ++k) {
        int d1 = j1+j2-j3-k, d2 = j1-m1-k, d3 = j2+m2-k, d4 = j3-j2+m1+k, d5 = j3-j1-m2+k;
        if (d1 < 0 || d2 < 0 || d3 < 0 || d4 < 0 || d5 < 0) continue;
        double den = factd(k)*factd(d1)*factd(d2)*factd(d3)*factd(d4)*factd(d5);
        s += ((k & 1) ? -1.0 : 1.0) / den;
    }
    return pref * s;
}

__device__ void qmat(int l, cdbl* q) {   // (2l+1)x(2l+1), row-major; real<->complex SH basis
    int n = 2*l + 1;
    for (int i = 0; i < n*n; ++i) { q[i].re = 0.0; q[i].im = 0.0; }
    const double is2 = 0.70710678118654752440;
    for (int m = -l; m < 0; ++m) {
        q[(l+m)*n + (l-m)].re = is2;     // col l+|m|
        q[(l+m)*n + (l+m)].im = -is2;    // col l-|m|
    }
    q[l*n + l].re = 1.0;
    for (int m = 1; m <= l; ++m) {
        double sgn = (m & 1) ? -1.0 : 1.0;
        q[(l+m)*n + (l+m)].re = sgn * is2;
        q[(l+m)*n + (l-m)].im = sgn * is2;
    }
    cdbl ph;   // (-i)^l
    if (l == 0)      { ph.re = 1.0;  ph.im = 0.0; }
    else if (l == 1) { ph.re = 0.0;  ph.im = -1.0; }
    else             { ph.re = -1.0; ph.im = 0.0; }
    for (int i = 0; i < n*n; ++i) q[i] = cmul(q[i], ph);
}

// cgb: dense 9x81 matrix; row ro = cat(o,mo), col = cat(i,mi)*9 + cat(f,mf)
__global__ __launch_bounds__(256) void cg_kernel(float* __restrict__ cgb) {
    for (int t = threadIdx.x; t < 729; t += blockDim.x) cgb[t] = 0.0f;
    __syncthreads();
    int p = threadIdx.x;
    if (p >= 15) return;
    int li = PATH_I[p], lf = PATH_F[p], lo = PATH_O[p];
    int ni = 2*li+1, nf = 2*lf+1, no = 2*lo+1;
    cdbl Qi[25], Qf[25], Qo[25];
    qmat(li, Qi); qmat(lf, Qf); qmat(lo, Qo);
    const int offs[3] = {0, 1, 4};
    for (int mo = 0; mo < no; ++mo)
    for (int mi = 0; mi < ni; ++mi)
    for (int mf = 0; mf < nf; ++mf) {
        double ar = 0.0;
        for (int ii = 0; ii < ni; ++ii)
        for (int kk = 0; kk < nf; ++kk) {
            int m1 = ii - li, m2 = kk - lf, m3 = m1 + m2;
            if (m3 < -lo || m3 > lo) continue;
            double c = su2_elem(li, lf, lo, m1, m2);
            if (c == 0.0) continue;
            cdbl t = cmul(Qi[ii*ni + mi], Qf[kk*nf + mf]);
            cdbl qo = Qo[(lo+m3)*no + mo]; qo.im = -qo.im;   // conj(Q3[n,m])
            t = cmul(t, qo);
            ar += t.re * c;
        }
        cgb[(offs[lo]+mo)*81 + (offs[li]+mi)*9 + (offs[lf]+mf)] = (float)ar;
    }
}

// =====================================================================
// Math helpers
// =====================================================================
__device__ __forceinline__ float eta_f(float x) {   // softplus(x) - log(2)
    float sp = fmaxf(x, 0.0f) + log1pf(expf(-fabsf(x)));
    return sp - 0.69314718055994530942f;
}

// =====================================================================
// Self-interaction:  U[a,d,j] = sum_c X[a,c,j] * W[c,d]  (+ b[d] at j==0)
// M = A*9 rows (a,j), K = Cin, N = Cout.  Uses V_WMMA_F32_16X16X4_F32.
// A-fragment addressing hoisted to (base,stride); W staged zero-padded
// in LDS so the K-loop has no divergent branches.
// =====================================================================
__device__ __forceinline__ void si_store(v8f acc, int row0, int nt, int lane,
                                         int Mtot, int Cout, const float* b, float* Uo) {
#pragma unroll
    for (int v = 0; v < 8; ++v) {
        int rt  = (lane < 16) ? v : v + 8;       // C/D layout: lanes 16-31 hold M=v+8
        int rr  = row0 + rt;
        int col = nt*16 + (lane & 15);
        if (rr < Mtot && col < Cout) {
            int aa = rr / 9, jj = rr - aa*9;
            float val = acc[v];
            if (jj == 0) val += b[col];
            Uo[((size_t)aa*Cout + col)*9 + jj] = val;
        }
    }
}

__global__ __launch_bounds__(256) void si_wmma_kernel(
    const float* __restrict__ Vc,                         // cat input [A][Cin][9] or null
    const float* __restrict__ V0, const float* __restrict__ V1, const float* __restrict__ V2,
    const float* __restrict__ scale,                      // per-(a,j) inverse norm or null
    const float* __restrict__ W,  const float* __restrict__ b,
    float* __restrict__ Uo, int A, int Cin, int Cout)
{
    __shared__ float Wl[24*32];                 // zero-padded W: [Kpad<=24][32]
    int nK   = (Cin + 3) >> 2;
    int Kpad = nK * 4;
    for (int t = threadIdx.x; t < Kpad*32; t += blockDim.x) {
        int k = t >> 5, n = t & 31;
        Wl[t] = (k < Cin && n < Cout) ? W[(size_t)k*Cout + n] : 0.0f;
    }
    __syncthreads();

    int lane = threadIdx.x & 31;
    int wave = (int)((blockIdx.x * blockDim.x + threadIdx.x) >> 5);
    int Mtot = A * 9;
    int mtiles = (Mtot + 15) >> 4;
    if (wave >= mtiles) return;

    int row0 = wave * 16;
    int r = row0 + (lane & 15);
    bool rvalid = r < Mtot;
    int rc = rvalid ? r : (Mtot - 1);
    int a = rc / 9, j = rc - a*9;
    float sc = rvalid ? 1.0f : 0.0f;            // pad rows contribute zero
    if (scale) sc = rvalid ? scale[rc] : 0.0f;

    // All input modes are affine in k: addr = Ab + k*As
    const float* Ab; int As;
    if (Vc)          { Ab = Vc + ((size_t)a*Cin)*9 + j;       As = 9; }
    else if (j == 0) { Ab = V0 + (size_t)a*Cin;               As = 1; }
    else if (j < 4)  { Ab = V1 + ((size_t)a*Cin)*3 + (j-1);   As = 3; }
    else             { Ab = V2 + ((size_t)a*Cin)*5 + (j-4);   As = 5; }

    int ntiles = (Cout + 15) >> 4;              // 1 or 2
    int khalf  = (lane >> 4) * 2;               // A layout: lanes 16-31 hold K=2,3
    int nc     = lane & 15;

    v8f acc0 = {};
    v8f acc1 = {};
    for (int kc = 0; kc < nK; ++kc) {
        int k0 = kc*4 + khalf;
        int k1 = k0 + 1;
        int k0c = (k0 < Cin) ? k0 : (Cin - 1);  // clamp address, mask value
        int k1c = (k1 < Cin) ? k1 : (Cin - 1);
        float a0 = Ab[(size_t)k0c * As];
        float a1 = Ab[(size_t)k1c * As];
        v2f av;
        av.x = (k0 < Cin) ? a0 * sc : 0.0f;
        av.y = (k1 < Cin) ? a1 * sc : 0.0f;
        v2f bv0;
        bv0.x = Wl[k0*32 + nc];
        bv0.y = Wl[k1*32 + nc];
        acc0 = __builtin_amdgcn_wmma_f32_16x16x4_f32(false, av, false, bv0, (short)0, acc0, false, false);
        if (ntiles > 1) {
            v2f bv1;
            bv1.x = Wl[k0*32 + 16 + nc];
            bv1.y = Wl[k1*32 + 16 + nc];
            acc1 = __builtin_amdgcn_wmma_f32_16x16x4_f32(false, av, false, bv1, (short)0, acc1, false, false);
        }
    }
    si_store(acc0, row0, 0, lane, Mtot, Cout, b, Uo);
    if (ntiles > 1) si_store(acc1, row0, 1, lane, Mtot, Cout, b, Uo);
}

// =====================================================================
// Radial MLP (12 -> 12 relu -> C) and angular harmonics Y[9] per (a,l)
// =====================================================================
__global__ __launch_bounds__(256) void radang_kernel(
    const float* __restrict__ rads, const float* __restrict__ vecs,
    const float* __restrict__ W1, const float* __restrict__ b1,
    const float* __restrict__ W2, const float* __restrict__ b2,
    float* __restrict__ R, float* __restrict__ Y, int AL, int C, int computeY)
{
    int t = blockIdx.x * blockDim.x + threadIdx.x;
    if (t >= AL) return;
    float x[12], h[12];
#pragma unroll
    for (int i = 0; i < 12; ++i) x[i] = rads[(size_t)t*12 + i];
#pragma unroll
    for (int jj = 0; jj < 12; ++jj) h[jj] = b1[jj];
#pragma unroll
    for (int i = 0; i < 12; ++i) {
        float xi = x[i];
#pragma unroll
        for (int jj = 0; jj < 12; ++jj) h[jj] += xi * W1[i*12 + jj];
    }
#pragma unroll
    for (int jj = 0; jj < 12; ++jj) h[jj] = fmaxf(h[jj], 0.0f);
    for (int c = 0; c < C; ++c) {
        float acc = b2[c];
#pragma unroll
        for (int jj = 0; jj < 12; ++jj) acc += h[jj] * W2[jj*C + c];
        R[(size_t)t*C + c] = acc;
    }
    if (computeY) {
        float vx = vecs[(size_t)t*3 + 0], vy = vecs[(size_t)t*3 + 1], vz = vecs[(size_t)t*3 + 2];
        float r2 = fmaxf(vx*vx + vy*vy + vz*vz, 1e-9f);
        float inv = 1.0f / r2;
        float* yo = Y + (size_t)t*9;
        yo[0] = 1.0f;
        yo[1] = vx; yo[2] = vy; yo[3] = vz;
        yo[4] = vx*vy*inv;
        yo[5] = vy*vz*inv;
        yo[6] = (-vx*vx - vy*vy + 2.0f*vz*vz) * (inv * 0.28867513459481288225f); // 1/(2*sqrt(3))
        yo[7] = vz*vx*inv;
        yo[8] = (vx*vx - vy*vy) * (0.5f*inv);
    }
}

// =====================================================================
// Convolution: one thread per (a,c).
// P[9x9] += (R * U_cat9) (x) Y_cat9 over 24 neighbors, then O9 = CG(9x81) @ P
// =====================================================================
__global__ __launch_bounds__(256) void conv_kernel(
    const float* __restrict__ U, const float* __restrict__ R,
    const float* __restrict__ Y, const int* __restrict__ idx,
    const float* __restrict__ cg, float* __restrict__ Out,
    int A, int C, int L)
{
    int t = blockIdx.x * blockDim.x + threadIdx.x;
    if (t >= A*C) return;
    int a = t / C, c = t - a*C;
    float P[81];
#pragma unroll
    for (int i = 0; i < 81; ++i) P[i] = 0.0f;

    const int* ip = idx + (size_t)a*L;
    for (int l = 0; l < L; ++l) {
        int pair = a*L + l;
        int n = ip[l];
        if (l + 1 < L) {
            int nn = ip[l+1];
            __builtin_prefetch(U + ((size_t)nn*C + c)*9, 0, 1);
        }
        float s = R[(size_t)pair*C + c];
        const float* yp = Y + (size_t)pair*9;
        const float* up = U + ((size_t)n*C + c)*9;
        float y[9], u[9];
#pragma unroll
        for (int jj = 0; jj < 9; ++jj) y[jj] = yp[jj];
#pragma unroll
        for (int jj = 0; jj < 9; ++jj) u[jj] = up[jj];
#pragma unroll
        for (int ic = 0; ic < 9; ++ic) {
            float su = s * u[ic];
#pragma unroll
            for (int fc = 0; fc < 9; ++fc) P[ic*9 + fc] += su * y[fc];
        }
    }
    float* op = Out + (size_t)t*9;
#pragma unroll
    for (int ro = 0; ro < 9; ++ro) {
        float accv = 0.0f;
#pragma unroll
        for (int col = 0; col < 81; ++col) accv += cg[ro*81 + col] * P[col];
        op[ro] = accv;
    }
}

// =====================================================================
// Per-(a,j) inverse channel norm:  1 / max(||X[a,:,j]||, 1e-9)
// =====================================================================
__global__ __launch_bounds__(256) void normfac_kernel(
    const float* __restrict__ X, float* __restrict__ N, int A, int C)
{
    int t = blockIdx.x * blockDim.x + threadIdx.x;
    if (t >= A*9) return;
    int a = t / 9, j = t - a*9;
    float s = 0.0f;
    for (int c = 0; c < C; ++c) {
        float v = X[((size_t)a*C + c)*9 + j];
        s += v*v;
    }
    N[t] = 1.0f / fmaxf(sqrtf(s), 1e-9f);
}

// =====================================================================
// Gated nonlinearity (TFN)
// =====================================================================
__global__ __launch_bounds__(256) void nonlin_kernel(
    const float* __restrict__ X, const float* __restrict__ b1,
    const float* __restrict__ b2, float* __restrict__ Out, int A, int C)
{
    int t = blockIdx.x * blockDim.x + threadIdx.x;
    if (t >= A*C) return;
    int c = t % C;
    const float* x = X + (size_t)t*9;
    float v[9];
#pragma unroll
    for (int jj = 0; jj < 9; ++jj) v[jj] = x[jj];
    float o0 = eta_f(v[0]);
    float t1 = sqrtf(v[1]*v[1] + v[2]*v[2] + v[3]*v[3] + 1e-9f) + b1[c];
    float s1 = eta_f(t1);
    float t2 = sqrtf(v[4]*v[4] + v[5]*v[5] + v[6]*v[6] + v[7]*v[7] + v[8]*v[8] + 1e-9f) + b2[c];
    float s2 = eta_f(t2);
    float* o = Out + (size_t)t*9;
    o[0] = o0;
    o[1] = v[1]*s1; o[2] = v[2]*s1; o[3] = v[3]*s1;
    o[4] = v[4]*s2; o[5] = v[5]*s2; o[6] = v[6]*s2; o[7] = v[7]*s2; o[8] = v[8]*s2;
}

// =====================================================================
// Final: channel-mean over atoms of l=0 part, then dense 4->4(elu)->256->1
// =====================================================================
__global__ __launch_bounds__(256) void final_kernel(
    const float* __restrict__ V, const float* __restrict__ W1, const float* __restrict__ b1,
    const float* __restrict__ W2, const float* __restrict__ b2,
    const float* __restrict__ W3, const float* __restrict__ b3,
    float* __restrict__ out, int A, int C)
{
    __shared__ float sm[4*256];
    __shared__ float E[4];
    __shared__ float h1[4];
    __shared__ float red[256];
    float acc[4] = {0.f, 0.f, 0.f, 0.f};
    for (int a = threadIdx.x; a < A; a += 256) {
#pragma unroll
        for (int c = 0; c < 4; ++c) acc[c] += V[((size_t)a*C + c)*9];
    }
#pragma unroll
    for (int c = 0; c < 4; ++c) sm[c*256 + threadIdx.x] = acc[c];
    __syncthreads();
    for (int s = 128; s > 0; s >>= 1) {
        if (threadIdx.x < s) {
#pragma unroll
            for (int c = 0; c < 4; ++c) sm[c*256 + threadIdx.x] += sm[c*256 + threadIdx.x + s];
        }
        __syncthreads();
    }
    if (threadIdx.x < 4) E[threadIdx.x] = sm[threadIdx.x*256] / (float)A;
    __syncthreads();
    if (threadIdx.x < 4) {
        float h = b1[threadIdx.x];
#pragma unroll
        for (int c = 0; c < 4; ++c) h += E[c] * W1[c*4 + threadIdx.x];
        h1[threadIdx.x] = (h > 0.0f) ? h : expm1f(h);   // elu
    }
    __syncthreads();
    float h2 = b2[threadIdx.x];
#pragma unroll
    for (int d = 0; d < 4; ++d) h2 += h1[d] * W2[d*256 + threadIdx.x];
    red[threadIdx.x] = h2 * W3[threadIdx.x];
    __syncthreads();
    for (int s = 128; s > 0; s >>= 1) {
        if (threadIdx.x < s) red[threadIdx.x] += red[threadIdx.x + s];
        __syncthreads();
    }
    if (threadIdx.x == 0) out[0] = red[0] + b3[0];
}

// =====================================================================
// Host orchestration
// =====================================================================
static inline int cdiv_h(int a, int b) { return (a + b - 1) / b; }

extern "C" void kernel_launch(void* const* d_in, const int* in_sizes, int n_in,
                              void* d_out, int out_size, void* d_ws, size_t ws_size,
                              hipStream_t stream)
{
    (void)n_in; (void)out_size; (void)ws_size;
    const float* V0   = (const float*)d_in[0];   // [A,3,1]
    const float* V1   = (const float*)d_in[1];   // [A,3,3]
    const float* V2   = (const float*)d_in[2];   // [A,3,5]
    const float* rads = (const float*)d_in[3];   // [A,L,12]
    const float* vecs = (const float*)d_in[4];   // [A,L,3]
    const int*   idxs = (const int*)d_in[5];     // [A,L]

    int A = in_sizes[0] / 3;
    int L = in_sizes[5] / A;
    int AL = A * L;

    // params flattened in setup_inputs() insertion order:
    // blocks m1/m2/m3 at 6/16/26 with {si1_W,si1_b,rW1,rb1,rW2,rb2,si2_W,si2_b,nl_b1,nl_b2}
    // dense at 36 with {W1,b1,W2,b2,W3,b3}
    int c1 = in_sizes[7], c2 = in_sizes[17], c3 = in_sizes[27];
    int cins[3]  = {3, c1, c2};
    int couts[3] = {c1, c2, c3};

    float* ws = (float*)d_ws;
    size_t off = 0;
    float* cgb = ws + off; off += 1024;
    float* Yb  = ws + off; off += (size_t)AL * 9;
    float* Rb  = ws + off; off += (size_t)AL * 24;
    float* Ub  = ws + off; off += (size_t)A * 24 * 9;
    float* Cb  = ws + off; off += (size_t)A * 24 * 9;
    float* Nb  = ws + off; off += (size_t)A * 9;
    float* Vb  = ws + off; off += (size_t)A * 24 * 9;

    cg_kernel<<<1, 256, 0, stream>>>(cgb);

    for (int blk = 0; blk < 3; ++blk) {
        int base = 6 + 10*blk;
        int Cin = cins[blk], Cout = couts[blk];
        const float* si1_W = (const float*)d_in[base + 0];
        const float* si1_b = (const float*)d_in[base + 1];
        const float* rW1   = (const float*)d_in[base + 2];
        const float* rb1   = (const float*)d_in[base + 3];
        const float* rW2   = (const float*)d_in[base + 4];
        const float* rb2   = (const float*)d_in[base + 5];
        const float* si2_W = (const float*)d_in[base + 6];
        const float* si2_b = (const float*)d_in[base + 7];
        const float* nl_b1 = (const float*)d_in[base + 8];
        const float* nl_b2 = (const float*)d_in[base + 9];

        int mtiles = cdiv_h(A*9, 16);
        int si_blocks = cdiv_h(mtiles, 8);     // 8 waves (wave32) per 256-thread block

        // SI1: block input -> U [A][Cout][9]
        si_wmma_kernel<<<si_blocks, 256, 0, stream>>>(
            (blk == 0) ? nullptr : Vb, V0, V1, V2, nullptr,
            si1_W, si1_b, Ub, A, Cin, Cout);

        // Radial MLP (+ angular Y, once)
        radang_kernel<<<cdiv_h(AL, 256), 256, 0, stream>>>(
            rads, vecs, rW1, rb1, rW2, rb2, Rb, Yb, AL, Cout, (blk == 0) ? 1 : 0);

        // Convolution over neighbors
        conv_kernel<<<cdiv_h(A*Cout, 256), 256, 0, stream>>>(
            Ub, Rb, Yb, idxs, cgb, Cb, A, Cout, L);

        // Channel-norm factors
        normfac_kernel<<<cdiv_h(A*9, 256), 256, 0, stream>>>(Cb, Nb, A, Cout);

        // SI2 with fused normalization: Cb -> Ub
        si_wmma_kernel<<<si_blocks, 256, 0, stream>>>(
            Cb, nullptr, nullptr, nullptr, Nb,
            si2_W, si2_b, Ub, A, Cout, Cout);

        // Gated nonlinearity: Ub -> Vb (next block input)
        nonlin_kernel<<<cdiv_h(A*Cout, 256), 256, 0, stream>>>(Ub, nl_b1, nl_b2, Vb, A, Cout);
    }

    final_kernel<<<1, 256, 0, stream>>>(
        Vb,
        (const float*)d_in[36], (const float*)d_in[37],
        (const float*)d_in[38], (const float*)d_in[39],
        (const float*)d_in[40], (const float*)d_in[41],
        (float*)d_out, A, couts[2]);
}